// VMModule_84275848282428
// MI455X (gfx1250) — compile-verified
//
#include <hip/hip_runtime.h>
#include <cstdint>

#define GSZ 300
#define CH  16

#ifndef __has_builtin
#define __has_builtin(x) 0
#endif

#if __has_builtin(__builtin_amdgcn_global_load_async_to_lds_b128)
#define HAVE_ASYNC_LDS 1
#else
#define HAVE_ASYNC_LDS 0
#endif

typedef int v4i_t __attribute__((ext_vector_type(4)));
typedef __attribute__((address_space(1))) v4i_t* gv4i_p;
typedef __attribute__((address_space(3))) v4i_t* lv4i_p;

__device__ __forceinline__ void wait_asynccnt0() {
#if __has_builtin(__builtin_amdgcn_s_wait_asynccnt)
  __builtin_amdgcn_s_wait_asynccnt(0);
#else
  asm volatile("s_wait_asynccnt 0" ::: "memory");
#endif
}

// ---------------------------------------------------------------------------
// Kernel 1: layout transform.
//   planes (3,C,G,G) -> planesT (3,G,G,C)   (corner gather = 64B contiguous)
//   lines  (3,C,G)   -> linesT  (3,G,C)
// ---------------------------------------------------------------------------
__global__ __launch_bounds__(256) void transpose_kernel(
    const float* __restrict__ planes, const float* __restrict__ lines,
    float* __restrict__ planesT, float* __restrict__ linesT) {
  const int plane_cells = 3 * GSZ * GSZ;  // 270000
  int idx = blockIdx.x * blockDim.x + threadIdx.x;
  if (idx < plane_cells) {
    int i  = idx / (GSZ * GSZ);
    int yx = idx % (GSZ * GSZ);
    const float* src = planes + (size_t)i * CH * GSZ * GSZ + yx;
    float v[CH];
#pragma unroll
    for (int c = 0; c < CH; ++c) v[c] = src[(size_t)c * GSZ * GSZ];
    float4* dst = (float4*)(planesT + (size_t)idx * CH);
#pragma unroll
    for (int q = 0; q < 4; ++q)
      dst[q] = make_float4(v[4 * q], v[4 * q + 1], v[4 * q + 2], v[4 * q + 3]);
  } else if (idx < plane_cells + 3 * GSZ) {
    int j = idx - plane_cells;            // j = i*G + y
    int i = j / GSZ;
    int y = j % GSZ;
    const float* src = lines + (size_t)i * CH * GSZ + y;
    float* dst = linesT + (size_t)j * CH;
#pragma unroll
    for (int c = 0; c < CH; ++c) dst[c] = src[(size_t)c * GSZ];
  }
}

// ---------------------------------------------------------------------------
// Kernel 2: main VM sampling. One thread per point.
// ---------------------------------------------------------------------------
__device__ __forceinline__ void lerp_idx(float coord, int& i0, int& i1, float& w) {
  float x  = (coord + 1.0f) * 0.5f * (float)(GSZ - 1);
  float xf = floorf(x);
  if (xf < 0.0f) xf = 0.0f;
  if (xf > (float)(GSZ - 1)) xf = (float)(GSZ - 1);
  i0 = (int)xf;
  i1 = min(i0 + 1, GSZ - 1);
  w  = x - xf;
}

__global__ __launch_bounds__(256) void vm_sample_kernel(
    const float* __restrict__ xyz, const float* __restrict__ planesT,
    const float* __restrict__ linesT, float* __restrict__ out, int n) {
  __shared__ float s_lines[3 * GSZ * CH];  // 14400 floats = 57.6 KB

  // Stage transposed lines into LDS via the gfx1250 async path.
  const int nchunks = (3 * GSZ * CH) / 4;  // 3600 x float4
  for (int j = threadIdx.x; j < nchunks; j += blockDim.x) {
#if HAVE_ASYNC_LDS
    __builtin_amdgcn_global_load_async_to_lds_b128(
        (gv4i_p)(v4i_t*)(void*)(linesT + (size_t)j * 4),
        (lv4i_p)(v4i_t*)(void*)(&s_lines[j * 4]),
        0, 0);
#else
    ((float4*)s_lines)[j] = ((const float4*)linesT)[j];
#endif
  }
#if HAVE_ASYNC_LDS
  wait_asynccnt0();
#endif
  __syncthreads();

  int nidx = blockIdx.x * blockDim.x + threadIdx.x;
  if (nidx >= n) return;

  const float cx = xyz[(size_t)nidx * 3 + 0];
  const float cy = xyz[(size_t)nidx * 3 + 1];
  const float cz = xyz[(size_t)nidx * 3 + 2];

  // matMode = ((1,2),(0,2),(0,1)); gx indexes W (fast dim), gy indexes H.
  const float gxs[3] = {cy, cx, cx};
  const float gys[3] = {cz, cz, cy};
  const float gls[3] = {cx, cy, cz};

#pragma unroll
  for (int i = 0; i < 3; ++i) {
    int x0, x1, y0, y1, l0i, l1i;
    float wx, wy, wl;
    lerp_idx(gxs[i], x0, x1, wx);
    lerp_idx(gys[i], y0, y1, wy);
    lerp_idx(gls[i], l0i, l1i, wl);

    const size_t pbase = (size_t)i * GSZ * GSZ * CH;
    const float4* r00 = (const float4*)(planesT + pbase + ((size_t)y0 * GSZ + x0) * CH);
    const float4* r01 = (const float4*)(planesT + pbase + ((size_t)y0 * GSZ + x1) * CH);
    const float4* r10 = (const float4*)(planesT + pbase + ((size_t)y1 * GSZ + x0) * CH);
    const float4* r11 = (const float4*)(planesT + pbase + ((size_t)y1 * GSZ + x1) * CH);
    const float4* la  = (const float4*)(&s_lines[(i * GSZ + l0i) * CH]);
    const float4* lb  = (const float4*)(&s_lines[(i * GSZ + l1i) * CH]);

    const float omwx = 1.0f - wx, omwy = 1.0f - wy, omwl = 1.0f - wl;
    float* outp = out + (size_t)i * CH * (size_t)n + (size_t)nidx;

#pragma unroll
    for (int q = 0; q < 4; ++q) {
      float4 v00 = r00[q], v01 = r01[q], v10 = r10[q], v11 = r11[q];
      float4 a = la[q], b = lb[q];

      float p0 = (v00.x * omwx + v01.x * wx) * omwy + (v10.x * omwx + v11.x * wx) * wy;
      float p1 = (v00.y * omwx + v01.y * wx) * omwy + (v10.y * omwx + v11.y * wx) * wy;
      float p2 = (v00.z * omwx + v01.z * wx) * omwy + (v10.z * omwx + v11.z * wx) * wy;
      float p3 = (v00.w * omwx + v01.w * wx) * omwy + (v10.w * omwx + v11.w * wx) * wy;

      float f0 = p0 * (a.x * omwl + b.x * wl);
      float f1 = p1 * (a.y * omwl + b.y * wl);
      float f2 = p2 * (a.z * omwl + b.z * wl);
      float f3 = p3 * (a.w * omwl + b.w * wl);

      // Streaming output (never re-read): non-temporal so the L2-resident
      // planes table isn't evicted by 384 MB of result traffic.
      __builtin_nontemporal_store(f0, outp + (size_t)(4 * q + 0) * n);
      __builtin_nontemporal_store(f1, outp + (size_t)(4 * q + 1) * n);
      __builtin_nontemporal_store(f2, outp + (size_t)(4 * q + 2) * n);
      __builtin_nontemporal_store(f3, outp + (size_t)(4 * q + 3) * n);
    }
  }
}

extern "C" void kernel_launch(void* const* d_in, const int* in_sizes, int n_in,
                              void* d_out, int out_size, void* d_ws, size_t ws_size,
                              hipStream_t stream) {
  (void)n_in; (void)out_size; (void)ws_size;
  const float* xyz    = (const float*)d_in[0];
  const float* planes = (const float*)d_in[1];
  const float* lines  = (const float*)d_in[2];
  float* outp = (float*)d_out;

  float* planesT = (float*)d_ws;                               // 17.28 MB
  float* linesT  = planesT + (size_t)3 * CH * GSZ * GSZ;       // +57.6 KB

  const int n = in_sizes[0] / 3;

  const int t_items = 3 * GSZ * GSZ + 3 * GSZ;
  transpose_kernel<<<(t_items + 255) / 256, 256, 0, stream>>>(planes, lines,
                                                              planesT, linesT);
  vm_sample_kernel<<<(n + 255) / 256, 256, 0, stream>>>(xyz, planesT, linesT,
                                                        outp, n);
}